// ModulationFilterbank_4148938408278
// MI455X (gfx1250) — compile-verified
//
#include <hip/hip_runtime.h>
#include <math.h>

// ---------------- problem constants ----------------
#define T_LEN 48000
#define N_ROWS 124          // 4*31
#define SB 192              // superblock samples (multiple of 16)
#define NSB 250             // T_LEN / SB
#define TILES 12            // SB / 16
#define NBP 7               // bandpass resonators
#define NCH 8               // 1 LP + 7 BP output channels

typedef __attribute__((ext_vector_type(2))) float v2f;
typedef __attribute__((ext_vector_type(8))) float v8f;

struct FBParams {
    float blp0, blp1, blp2, alp1, alp2;       // Butterworth biquad
    float m00, m01, m10, m11;                 // M^SB (LP homogeneous transition)
    float b0[NBP];                            // 1 - e0
    float pr[NBP], pi[NBP];                   // complex pole
    float psr[NBP], psi[NBP];                 // pole^SB
};

// ---------------------------------------------------------------------------
// Pass 1: per (row, superblock) run all filters from zero state; store final
// local state (carry). 16 floats per (row, sb): [s0,s1, (yr,yi)*7]
// ---------------------------------------------------------------------------
__global__ void mfb_pass1(const float* __restrict__ x,
                          float* __restrict__ carryLocal, FBParams P) {
    int tid = blockIdx.x * blockDim.x + threadIdx.x;
    if (tid >= N_ROWS * NSB) return;
    int n = tid / NSB, sb = tid % NSB;

    const float4* xp = (const float4*)(x + (size_t)n * T_LEN + (size_t)sb * SB);
    float s0 = 0.f, s1 = 0.f;
    float yr[NBP], yi[NBP];
#pragma unroll
    for (int m = 0; m < NBP; ++m) { yr[m] = 0.f; yi[m] = 0.f; }

    for (int q = 0; q < SB / 4; ++q) {
        float4 xv = xp[q];
        float xs[4] = {xv.x, xv.y, xv.z, xv.w};
#pragma unroll
        for (int i = 0; i < 4; ++i) {
            float xt = xs[i];
            float y = P.blp0 * xt + s0;
            s0 = P.blp1 * xt - P.alp1 * y + s1;
            s1 = P.blp2 * xt - P.alp2 * y;
#pragma unroll
            for (int m = 0; m < NBP; ++m) {
                float nr = P.b0[m] * xt + P.pr[m] * yr[m] - P.pi[m] * yi[m];
                float ni =               P.pi[m] * yr[m] + P.pr[m] * yi[m];
                yr[m] = nr; yi[m] = ni;
            }
        }
    }
    float* c = carryLocal + ((size_t)sb * N_ROWS + n) * 16;
    c[0] = s0; c[1] = s1;
#pragma unroll
    for (int m = 0; m < NBP; ++m) { c[2 + 2 * m] = yr[m]; c[3 + 2 * m] = yi[m]; }
}

// ---------------------------------------------------------------------------
// Pass 2: per row, scan carries across superblocks with constant transitions.
// stateIn[sb] = true state just BEFORE superblock sb.
// ---------------------------------------------------------------------------
__global__ void mfb_pass2(const float* __restrict__ carryLocal,
                          float* __restrict__ stateIn, FBParams P) {
    int n = blockIdx.x * blockDim.x + threadIdx.x;
    if (n >= N_ROWS) return;
    float S[16];
#pragma unroll
    for (int i = 0; i < 16; ++i) S[i] = 0.f;

    for (int b = 0; b < NSB; ++b) {
        float* o = stateIn + ((size_t)b * N_ROWS + n) * 16;
        const float* c = carryLocal + ((size_t)b * N_ROWS + n) * 16;
#pragma unroll
        for (int i = 0; i < 16; ++i) o[i] = S[i];
        float t0 = S[0], t1 = S[1];
        S[0] = P.m00 * t0 + P.m01 * t1 + c[0];
        S[1] = P.m10 * t0 + P.m11 * t1 + c[1];
#pragma unroll
        for (int m = 0; m < NBP; ++m) {
            float ar = P.psr[m], ai = P.psi[m];
            float r = S[2 + 2 * m], im = S[3 + 2 * m];
            S[2 + 2 * m] = ar * r - ai * im + c[2 + 2 * m];
            S[3 + 2 * m] = ai * r + ar * im + c[3 + 2 * m];
        }
    }
}

// ---------------------------------------------------------------------------
// Pass 3a: LP output channel (exact scalar recompute, parallel over (row,sb))
// ---------------------------------------------------------------------------
__global__ void mfb_pass3_lp(const float* __restrict__ x,
                             const float* __restrict__ stateIn,
                             float* __restrict__ out, FBParams P) {
    int tid = blockIdx.x * blockDim.x + threadIdx.x;
    if (tid >= N_ROWS * NSB) return;
    int n = tid / NSB, sb = tid % NSB;

    const float* st = stateIn + ((size_t)sb * N_ROWS + n) * 16;
    float s0 = st[0], s1 = st[1];
    const float4* xp = (const float4*)(x + (size_t)n * T_LEN + (size_t)sb * SB);
    float4* op = (float4*)(out + (size_t)n * NCH * T_LEN + (size_t)sb * SB);

    for (int q = 0; q < SB / 4; ++q) {
        float4 xv = xp[q];
        float xs[4] = {xv.x, xv.y, xv.z, xv.w};
        float ys[4];
#pragma unroll
        for (int i = 0; i < 4; ++i) {
            float xt = xs[i];
            float y = P.blp0 * xt + s0;
            s0 = P.blp1 * xt - P.alp1 * y + s1;
            s1 = P.blp2 * xt - P.alp2 * y;
            ys[i] = 2.0f * y;
        }
        float4 ov = {ys[0], ys[1], ys[2], ys[3]};
        op[q] = ov;
    }
}

// ---------------------------------------------------------------------------
// Pass 3b: BP channels via WMMA. One wave per (rowgroup of 16, superblock).
// Per 16-sample tile: D[row][n] = sum_k X[row][k] * L^T[k][n]  (+ carry term),
// L[n][k] = b0 * p^(n-k) (lower triangular), complex -> 2 real WMMA chains.
// ---------------------------------------------------------------------------
__global__ void __launch_bounds__(256)
mfb_pass3_bp(const float* __restrict__ x, const float* __restrict__ stateIn,
             float* __restrict__ out, FBParams P) {
    int wv = blockIdx.x * (blockDim.x >> 5) + (threadIdx.x >> 5);   // 0..1999
    int g  = wv & 7;          // row group
    int sb = wv >> 3;         // superblock
    int lane = threadIdx.x & 31;
    int hn = lane & 15;       // A: row, B/D: column(sample)
    int hi = lane >> 4;       // half select

    int rowc = g * 16 + hn; if (rowc > N_ROWS - 1) rowc = N_ROWS - 1; // clamp loads
    const float* xrow_base = x + (size_t)rowc * T_LEN + (size_t)sb * SB;
    const float* st = stateIn + ((size_t)sb * N_ROWS + rowc) * 16;

#pragma unroll
    for (int m = 0; m < NBP; ++m) {
        // ---- per-lane B fragments (L^T) + p^(hn+1), via one power sweep ----
        float br_[4][2], bi_[4][2];
#pragma unroll
        for (int c = 0; c < 4; ++c) { br_[c][0] = br_[c][1] = 0.f; bi_[c][0] = bi_[c][1] = 0.f; }
        float pw1r = 0.f, pw1i = 0.f;
        {
            float r = 1.f, im = 0.f;
#pragma unroll
            for (int e = 0; e <= 16; ++e) {
#pragma unroll
                for (int c = 0; c < 4; ++c)
#pragma unroll
                    for (int j = 0; j < 2; ++j)
                        if (hn - (4 * c + hi * 2 + j) == e) {
                            br_[c][j] = P.b0[m] * r;
                            bi_[c][j] = P.b0[m] * im;
                        }
                if (hn + 1 == e) { pw1r = r; pw1i = im; }
                float t = r * P.pr[m] - im * P.pi[m];
                im = r * P.pi[m] + im * P.pr[m];
                r = t;
            }
        }
        // carry y[sb_start-1] for this lane's row (lane hn holds row g*16+hn)
        float cyr = st[2 + 2 * m], cyi = st[3 + 2 * m];

        for (int tl = 0; tl < TILES; ++tl) {
            const float* xrow = xrow_base + tl * 16;
            v8f dre = {0.f,0.f,0.f,0.f,0.f,0.f,0.f,0.f};
            v8f dim_ = {0.f,0.f,0.f,0.f,0.f,0.f,0.f,0.f};
#pragma unroll
            for (int c = 0; c < 4; ++c) {
                float2 a2 = *(const float2*)(xrow + c * 4 + hi * 2);
                v2f av; av.x = a2.x; av.y = a2.y;
                v2f bre; bre.x = br_[c][0]; bre.y = br_[c][1];
                v2f bim; bim.x = bi_[c][0]; bim.y = bi_[c][1];
                dre = __builtin_amdgcn_wmma_f32_16x16x4_f32(
                        false, av, false, bre, (short)0, dre, false, false);
                dim_ = __builtin_amdgcn_wmma_f32_16x16x4_f32(
                        false, av, false, bim, (short)0, dim_, false, false);
            }
            // add incoming-state contribution: + p^(n+1) * y_carry[row]
#pragma unroll
            for (int r = 0; r < 8; ++r) {
                int src = r + ((lane & 16) >> 1);        // carry of row r+hi*8
                float wr = __shfl(cyr, src);
                float wi = __shfl(cyi, src);
                dre[r] += pw1r * wr - pw1i * wi;
                dim_[r] += pw1r * wi + pw1i * wr;
            }
            // extract next carry: total y at sample 15 of each row
            float ncr = 0.f, nci = 0.f;
            int srcTop = 15 + ((hn >> 3) << 4);
#pragma unroll
            for (int r = 0; r < 8; ++r) {
                float tr = __shfl(dre[r], srcTop);
                float ti = __shfl(dim_[r], srcTop);
                if ((hn & 7) == r) { ncr = tr; nci = ti; }
            }
            cyr = ncr; cyi = nci;
            // stores: reg r -> row g*16+r+hi*8, sample sb*SB+tl*16+hn, ch 1+m
            int tidx = sb * SB + tl * 16 + hn;
#pragma unroll
            for (int r = 0; r < 8; ++r) {
                int rowo = g * 16 + r + hi * 8;
                if (rowo < N_ROWS) {
                    float v;
                    if (m < 2) v = 2.0f * dre[r];                       // keep phase
                    else       v = 2.0f * sqrtf(dre[r]*dre[r] + dim_[r]*dim_[r]);
                    out[((size_t)rowo * NCH + (1 + m)) * T_LEN + tidx] = v;
                }
            }
        }
    }
}

// ---------------------------------------------------------------------------
extern "C" void kernel_launch(void* const* d_in, const int* in_sizes, int n_in,
                              void* d_out, int out_size, void* d_ws, size_t ws_size,
                              hipStream_t stream) {
    const float* x = (const float*)d_in[0];
    float* out = (float*)d_out;
    float* carryLocal = (float*)d_ws;
    float* stateIn = carryLocal + (size_t)NSB * N_ROWS * 16;

    // ---- host-side coefficient computation (double precision) ----
    FBParams P;
    const double FS = 16000.0, Q = 2.0;
    {
        double K = tan(M_PI * 2.5 / FS);
        double nn = 1.0 / (1.0 + sqrt(2.0) * K + K * K);
        double b0 = K * K * nn, b1 = 2.0 * K * K * nn, b2 = K * K * nn;
        double a1 = 2.0 * (K * K - 1.0) * nn, a2 = (1.0 - sqrt(2.0) * K + K * K) * nn;
        P.blp0 = (float)b0; P.blp1 = (float)b1; P.blp2 = (float)b2;
        P.alp1 = (float)a1; P.alp2 = (float)a2;
        // M = [[-a1, 1], [-a2, 0]]  ->  M^SB
        double c00 = 1, c01 = 0, c10 = 0, c11 = 1;
        for (int i = 0; i < SB; ++i) {
            double n00 = -a1 * c00 + c10, n01 = -a1 * c01 + c11;
            double n10 = -a2 * c00,       n11 = -a2 * c01;
            c00 = n00; c01 = n01; c10 = n10; c11 = n11;
        }
        P.m00 = (float)c00; P.m01 = (float)c01; P.m10 = (float)c10; P.m11 = (float)c11;
    }
    {
        double mfc[NBP]; mfc[0] = 5.0; mfc[1] = 10.0;
        double rr = (1.0 + 1.0 / (2.0 * Q)) / (1.0 - 1.0 / (2.0 * Q));
        double f = 10.0 * rr; int cnt = 2;
        while (f <= 150.0 && cnt < NBP) { mfc[cnt++] = f; f *= rr; }
        for (int m = 0; m < NBP; ++m) {
            double bw = (mfc[m] <= 10.0) ? 5.0 : mfc[m] / Q;
            double e0 = exp(-M_PI * bw / FS);
            double th = 2.0 * M_PI * mfc[m] / FS;
            double prd = e0 * cos(th), pid = e0 * sin(th);
            P.b0[m] = (float)(1.0 - e0);
            P.pr[m] = (float)prd; P.pi[m] = (float)pid;
            double sr = 1.0, si = 0.0;
            for (int i = 0; i < SB; ++i) {
                double t = sr * prd - si * pid; si = sr * pid + si * prd; sr = t;
            }
            P.psr[m] = (float)sr; P.psi[m] = (float)si;
        }
    }

    int nthreads = N_ROWS * NSB;
    int blocks = (nthreads + 255) / 256;
    mfb_pass1<<<blocks, 256, 0, stream>>>(x, carryLocal, P);
    mfb_pass2<<<1, 128, 0, stream>>>(carryLocal, stateIn, P);
    mfb_pass3_lp<<<blocks, 256, 0, stream>>>(x, stateIn, out, P);
    mfb_pass3_bp<<<(8 * NSB) / 8, 256, 0, stream>>>(x, stateIn, out, P);
}